// Beamformor_89653147336805
// MI455X (gfx1250) — compile-verified
//
#include <hip/hip_runtime.h>

// MI455X (gfx1250) beamformer.
// Memory-bound: 303 MB total traffic / 23.3 TB/s ≈ 13 us floor.
// Strategy: stream input with NT hints (never reused, > L2), keep W (0.4 MB)
// cached, and do the per-frame complex reduction as diag(X * W^T) with
// chained v_wmma_f32_16x16x4_f32 — operands load straight into the WMMA
// register layout (no LDS staging), so the matrix path is free under the
// bandwidth roofline.

typedef float v2f __attribute__((ext_vector_type(2)));
typedef float v8f __attribute__((ext_vector_type(8)));

#define NUM_BEAM 24
#define NUM_BIN  257
#define NUM_CH   8
#define N_FRAMES 16384
#define FRAMES_PER_BLOCK 8          // 8 waves/block, one frame per wave
#define PLANE (NUM_BIN * NUM_CH)    // floats per (frame, component) plane: 2056

__global__ __launch_bounds__(256)
void Beamformor_89653147336805_kernel(const float* __restrict__ input,
                                      const float* __restrict__ W,
                                      const int*   __restrict__ beam_id,
                                      float*       __restrict__ out)
{
    const int lane = threadIdx.x & 31;
    const int wave = threadIdx.x >> 5;

    if (blockIdx.y == 16) {
        // ---- tail: bin 256 (257 = 16*16 + 1), plain VALU path ----
        if (blockIdx.x >= (N_FRAMES / 256)) return;
        const int t = blockIdx.x * 256 + threadIdx.x;
        const int b = beam_id[t];
        const float* x = input + (size_t)t * (2 * PLANE) + 256 * NUM_CH;
        const float* w = W     + (size_t)b * (2 * PLANE) + 256 * NUM_CH;
        float or_ = 0.f, oi_ = 0.f;
#pragma unroll
        for (int c = 0; c < NUM_CH; ++c) {
            const float xr = x[c], xi = x[PLANE + c];
            const float wr = w[c], wi = w[PLANE + c];
            or_ = fmaf(xr, wr, or_); or_ = fmaf(xi, wi, or_);
            oi_ = fmaf(xi, wr, oi_); oi_ = fmaf(-xr, wi, oi_);
        }
        const size_t o = (size_t)t * 2 * NUM_BIN + 256;
        __builtin_nontemporal_store(or_, out + o);
        __builtin_nontemporal_store(oi_, out + o + NUM_BIN);
        return;
    }

    // ---- WMMA path: wave = one frame, 16-bin window [f0, f0+16) ----
    const int t  = blockIdx.x * FRAMES_PER_BLOCK + wave;   // frame (wave-uniform)
    const int f0 = blockIdx.y * 16;                        // bin window start
    const int lm = lane & 15;                              // row/col index
    const int hiOff = (lane >> 4) * 2;                     // hi half-wave: K+2

    const int b = beam_id[t];                              // wave-uniform beam

    // Feature vector per bin: K = [xr ch0..7, xi ch0..7] (16 features).
    // 32-bit A 16x4 layout per chunk c: v0 = K=4c (lanes0-15) / 4c+2 (16-31),
    //                                   v1 = K=4c+1 / 4c+3  -> one b64 per lane.
    const float* xr = input + (size_t)t * (2 * PLANE) + (size_t)(f0 + lm) * NUM_CH;
    const float* xi = xr + PLANE;
    const float* wr = W     + (size_t)b * (2 * PLANE) + (size_t)(f0 + lm) * NUM_CH;
    const float* wi = wr + PLANE;

    // A = X tile (streamed once -> non-temporal)
    const v2f a0 = __builtin_nontemporal_load((const v2f*)(xr + hiOff));      // feats 0-3
    const v2f a1 = __builtin_nontemporal_load((const v2f*)(xr + hiOff + 4));  // feats 4-7
    const v2f a2 = __builtin_nontemporal_load((const v2f*)(xi + hiOff));      // feats 8-11
    const v2f a3 = __builtin_nontemporal_load((const v2f*)(xi + hiOff + 4));  // feats 12-15

    // B_r columns = [wr ; wi] per bin (hot table -> cached loads)
    const v2f br0 = *(const v2f*)(wr + hiOff);
    const v2f br1 = *(const v2f*)(wr + hiOff + 4);
    const v2f br2 = *(const v2f*)(wi + hiOff);
    const v2f br3 = *(const v2f*)(wi + hiOff + 4);

    // B_i columns = [-wi ; wr]  (register reuse + 4 negations)
    const v2f bi0 = -br2, bi1 = -br3, bi2 = br0, bi3 = br1;

    v8f dr = {};  // D_r: diag = out_r for bins f0..f0+15
    v8f di = {};  // D_i: diag = out_i
    // K=16 via 4 chained 16x16x4 fp32 WMMAs per output matrix.
    dr = __builtin_amdgcn_wmma_f32_16x16x4_f32(false, a0, false, br0, (short)0, dr, false, false);
    di = __builtin_amdgcn_wmma_f32_16x16x4_f32(false, a0, false, bi0, (short)0, di, false, false);
    dr = __builtin_amdgcn_wmma_f32_16x16x4_f32(false, a1, false, br1, (short)0, dr, false, false);
    di = __builtin_amdgcn_wmma_f32_16x16x4_f32(false, a1, false, bi1, (short)0, di, false, false);
    dr = __builtin_amdgcn_wmma_f32_16x16x4_f32(false, a2, false, br2, (short)0, dr, false, false);
    di = __builtin_amdgcn_wmma_f32_16x16x4_f32(false, a2, false, bi2, (short)0, di, false, false);
    dr = __builtin_amdgcn_wmma_f32_16x16x4_f32(false, a3, false, br3, (short)0, dr, false, false);
    di = __builtin_amdgcn_wmma_f32_16x16x4_f32(false, a3, false, bi3, (short)0, di, false, false);

    // Diagonal extraction. D layout: v_j holds (M=j, N=lane) on lanes 0-15 and
    // (M=j+8, N=lane-16) on lanes 16-31. diag(N) lives on lanes 0-7 (v_lane)
    // and lanes 24-31 (v_{lane-24}). 7-cndmask select tree per matrix:
    const int k = lane & 7;
    float r01 = (k & 1) ? dr[1] : dr[0];
    float r23 = (k & 1) ? dr[3] : dr[2];
    float r45 = (k & 1) ? dr[5] : dr[4];
    float r67 = (k & 1) ? dr[7] : dr[6];
    float i01 = (k & 1) ? di[1] : di[0];
    float i23 = (k & 1) ? di[3] : di[2];
    float i45 = (k & 1) ? di[5] : di[4];
    float i67 = (k & 1) ? di[7] : di[6];
    float r03 = (k & 2) ? r23 : r01;
    float r47 = (k & 2) ? r67 : r45;
    float i03 = (k & 2) ? i23 : i01;
    float i47 = (k & 2) ? i67 : i45;
    const float diag_r = (k & 4) ? r47 : r03;
    const float diag_i = (k & 4) ? i47 : i03;

    const bool lo8 = (lane < 8);
    const bool hi8 = (lane >= 24);
    if (lo8 | hi8) {                       // after all WMMAs: divergence OK
        const int n = lo8 ? lane : (lane - 16);      // bin within window
        const size_t o = (size_t)t * 2 * NUM_BIN + (f0 + n);
        __builtin_nontemporal_store(diag_r, out + o);            // out[t,0,f]
        __builtin_nontemporal_store(diag_i, out + o + NUM_BIN);  // out[t,1,f]
    }
}

extern "C" void kernel_launch(void* const* d_in, const int* in_sizes, int n_in,
                              void* d_out, int out_size, void* d_ws, size_t ws_size,
                              hipStream_t stream) {
    const float* input   = (const float*)d_in[0];
    const float* W       = (const float*)d_in[1];
    const int*   beam_id = (const int*)d_in[2];
    float*       out     = (float*)d_out;

    // grid.x: 16384/8 frame groups; grid.y: 16 WMMA bin-windows + 1 tail block row
    dim3 grid(N_FRAMES / FRAMES_PER_BLOCK, 17);
    Beamformor_89653147336805_kernel<<<grid, 256, 0, stream>>>(input, W, beam_id, out);
}